// FullyConnectedSteerableGeometricProductLayer_9139690406431
// MI455X (gfx1250) — compile-verified
//
#include <hip/hip_runtime.h>

typedef __attribute__((ext_vector_type(2))) float v2f;
typedef __attribute__((ext_vector_type(4))) float v4f;
typedef __attribute__((ext_vector_type(8))) float v8f;

#define SEQ 2048

// ---------------------------------------------------------------------------
// Kernel 1: per output row n, compute q[b,n,i] = sum_m x[b,m,i]*Wq[n,m,g(i)]
// (and same for k), normalize per-grade, then emit:
//   KN[b,n,i]   = normalized k
//   A[b,n,j,kc] = sum_i qn[i] * Cayley[i,j,kc]   (signed permutation scatter)
// One block per n (2048 blocks, 256 threads). Reads of Wq/Wk (128MB) dominate.
// ---------------------------------------------------------------------------
__global__ __launch_bounds__(256) void qk_prep_kernel(
    const float* __restrict__ x, const float* __restrict__ Wq,
    const float* __restrict__ Wk, const float* __restrict__ a,
    float* __restrict__ Aout, float* __restrict__ KNout)
{
  const int n = blockIdx.x;
  const int tid = threadIdx.x;

  float acc[32];
#pragma unroll
  for (int v = 0; v < 32; ++v) acc[v] = 0.f;

  const v4f* wq4 = (const v4f*)(Wq + (size_t)n * SEQ * 4);
  const v4f* wk4 = (const v4f*)(Wk + (size_t)n * SEQ * 4);
  const v4f* x4  = (const v4f*)x;

  for (int m = tid; m < SEQ; m += 256) {
    v4f wq  = wq4[m];
    v4f wk  = wk4[m];
    v4f x00 = x4[m * 2],           x01 = x4[m * 2 + 1];
    v4f x10 = x4[2 * SEQ + m * 2], x11 = x4[2 * SEQ + m * 2 + 1];
    // grade map per blade: {0,1,1,1,2,2,2,3} -> wq.x,.y,.y,.y,.z,.z,.z,.w
    acc[0]  += x00.x * wq.x; acc[1]  += x00.y * wq.y; acc[2]  += x00.z * wq.y; acc[3]  += x00.w * wq.y;
    acc[4]  += x01.x * wq.z; acc[5]  += x01.y * wq.z; acc[6]  += x01.z * wq.z; acc[7]  += x01.w * wq.w;
    acc[8]  += x10.x * wq.x; acc[9]  += x10.y * wq.y; acc[10] += x10.z * wq.y; acc[11] += x10.w * wq.y;
    acc[12] += x11.x * wq.z; acc[13] += x11.y * wq.z; acc[14] += x11.z * wq.z; acc[15] += x11.w * wq.w;
    acc[16] += x00.x * wk.x; acc[17] += x00.y * wk.y; acc[18] += x00.z * wk.y; acc[19] += x00.w * wk.y;
    acc[20] += x01.x * wk.z; acc[21] += x01.y * wk.z; acc[22] += x01.z * wk.z; acc[23] += x01.w * wk.w;
    acc[24] += x10.x * wk.x; acc[25] += x10.y * wk.y; acc[26] += x10.z * wk.y; acc[27] += x10.w * wk.y;
    acc[28] += x11.x * wk.z; acc[29] += x11.y * wk.z; acc[30] += x11.z * wk.z; acc[31] += x11.w * wk.w;
  }

  __shared__ float sm[256 * 33];  // pad 33 to stay bank-conflict free
#pragma unroll
  for (int v = 0; v < 32; ++v) sm[tid * 33 + v] = acc[v];
  __syncthreads();
  for (int stride = 128; stride > 0; stride >>= 1) {
    if (tid < stride) {
#pragma unroll
      for (int v = 0; v < 32; ++v) sm[tid * 33 + v] += sm[(tid + stride) * 33 + v];
    }
    __syncthreads();
  }

  if (tid < 2) {
    const int b = tid;
    float q[8], kq[8];
#pragma unroll
    for (int i = 0; i < 8; ++i) { q[i] = sm[b * 8 + i]; kq[i] = sm[16 + b * 8 + i]; }

    float sa[4];
#pragma unroll
    for (int g = 0; g < 4; ++g) sa[g] = 1.f / (1.f + expf(-a[n * 4 + g]));

#pragma unroll
    for (int pass = 0; pass < 2; ++pass) {
      float* v = pass ? kq : q;
      float n0 = fabsf(v[0]);
      float n1 = sqrtf(v[1] * v[1] + v[2] * v[2] + v[3] * v[3]);
      float n2 = sqrtf(v[4] * v[4] + v[5] * v[5] + v[6] * v[6]);
      float n3 = fabsf(v[7]);
      float d0 = sa[0] * (n0 - 1.f) + 1.f + 1e-6f;
      float d1 = sa[1] * (n1 - 1.f) + 1.f + 1e-6f;
      float d2 = sa[2] * (n2 - 1.f) + 1.f + 1e-6f;
      float d3 = sa[3] * (n3 - 1.f) + 1.f + 1e-6f;
      v[0] /= d0; v[1] /= d1; v[2] /= d1; v[3] /= d1;
      v[4] /= d2; v[5] /= d2; v[6] /= d2; v[7] /= d3;
    }

    float* knp = KNout + ((size_t)b * SEQ + n) * 8;
#pragma unroll
    for (int i = 0; i < 8; ++i) knp[i] = kq[i];

    // Cayley table Cl(3,0): blade_i * blade_kc = sg[i][kc] * blade_{jm[i][kc]}
    const int jm[64] = {
      0,1,2,3,4,5,6,7,
      1,0,4,5,2,3,7,6,
      2,4,0,6,1,7,3,5,
      3,5,6,0,7,1,2,4,
      4,2,1,7,0,6,5,3,
      5,3,7,1,6,0,4,2,
      6,7,3,2,5,4,0,1,
      7,6,5,4,3,2,1,0};
    const float sg[64] = {
      1, 1, 1, 1, 1, 1, 1, 1,
      1, 1, 1, 1, 1, 1, 1, 1,
      1,-1, 1, 1,-1,-1, 1,-1,
      1,-1,-1, 1, 1,-1,-1, 1,
      1,-1, 1, 1,-1,-1, 1,-1,
      1,-1,-1, 1, 1,-1,-1, 1,
      1, 1,-1, 1,-1, 1,-1,-1,
      1, 1,-1, 1,-1, 1,-1,-1};
    float Al[64];
#pragma unroll
    for (int z = 0; z < 64; ++z) Al[z] = 0.f;
#pragma unroll
    for (int i = 0; i < 8; ++i)
#pragma unroll
      for (int kc = 0; kc < 8; ++kc)
        Al[jm[i * 8 + kc] * 8 + kc] += sg[i * 8 + kc] * q[i];

    float* ap = Aout + ((size_t)b * SEQ + n) * 64;
#pragma unroll
    for (int z = 0; z < 64; ++z) ap[z] = Al[z];
  }
}

// ---------------------------------------------------------------------------
// Kernel 2: out[(s*8+j), t] = sum_k A[(s,j),k] * KN[t,k]   per batch b.
// GEMM M=16384, N=2048, K=8 -> two chained V_WMMA_F32_16X16X4_F32 per tile.
// Each wave: one (b, s-pair, quarter-of-t); A fragment (4 VGPRs) hoisted,
// inner loop = load 4x16 B fragments, 2 WMMAs, 1KB non-temporal store.
// ---------------------------------------------------------------------------
__global__ __launch_bounds__(256) void geoprod_wmma_kernel(
    const float* __restrict__ A, const float* __restrict__ KN,
    float* __restrict__ out)
{
  const int lane   = threadIdx.x & 31;
  const int wid    = blockIdx.x * 8 + (threadIdx.x >> 5);  // 0..8191
  const int tchunk = wid & 3;
  const int spair  = (wid >> 2) & 1023;
  const int b      = wid >> 12;
  const int s0     = spair * 2;
  const int m16    = lane & 15;   // M within 16x16 tile
  const int khalf  = lane >> 4;   // high lanes hold K+2 (A) / K+2 rows (B)

  const int sA = s0 + (m16 >> 3);
  const int jA = m16 & 7;

  // A fragment: 16x8 = two 16x4 chunks. VGPR0: K = khalf*2, VGPR1: K = khalf*2+1
  const float* arow = A + (((size_t)b * SEQ + sA) * 64) + jA * 8 + khalf * 2;
  v2f aLo, aHi;
  aLo.x = arow[0]; aLo.y = arow[1];   // K = {0,1} (lanes 0-15) / {2,3} (lanes 16-31)
  aHi.x = arow[4]; aHi.y = arow[5];   // K = {4,5} / {6,7}

  const int tstart = tchunk * 512;
  // B fragment source: KN[b, t, k]; VGPR0 lane -> (t0+m16, khalf*2), VGPR1 -> +1
  const float* kb = KN + ((size_t)b * SEQ + tstart + m16) * 8 + khalf * 2;
  // D store: lane -> out[b, s0+khalf, t0+m16, j=0..7]
  v4f* outp = (v4f*)(out + (((size_t)b * SEQ + (s0 + khalf)) * SEQ + tstart + m16) * 8);

  for (int it = 0; it < 32; ++it) {
    v2f bLo, bHi;
    bLo.x = kb[0]; bLo.y = kb[1];
    bHi.x = kb[4]; bHi.y = kb[5];
    kb += 16 * 8;

    v8f c = {};
    c = __builtin_amdgcn_wmma_f32_16x16x4_f32(false, aLo, false, bLo,
                                              (short)0, c, false, false);
    c = __builtin_amdgcn_wmma_f32_16x16x4_f32(false, aHi, false, bHi,
                                              (short)0, c, false, false);

    v4f lo = { c[0], c[1], c[2], c[3] };
    v4f hi = { c[4], c[5], c[6], c[7] };
    __builtin_nontemporal_store(lo, outp);       // 268MB output >> L2: bypass
    __builtin_nontemporal_store(hi, outp + 1);
    outp += 32;  // advance 16 t * 8 floats
  }
}

extern "C" void kernel_launch(void* const* d_in, const int* in_sizes, int n_in,
                              void* d_out, int out_size, void* d_ws, size_t ws_size,
                              hipStream_t stream) {
  const float* x  = (const float*)d_in[0];
  const float* Wq = (const float*)d_in[1];
  const float* Wk = (const float*)d_in[2];
  const float* a  = (const float*)d_in[3];
  float* out = (float*)d_out;

  float* Aws = (float*)d_ws;                 // 2*2048*64 floats = 1 MB
  float* KN  = Aws + (size_t)2 * SEQ * 64;   // 2*2048*8 floats  = 128 KB

  qk_prep_kernel<<<SEQ, 256, 0, stream>>>(x, Wq, Wk, a, Aws, KN);
  geoprod_wmma_kernel<<<1024, 256, 0, stream>>>(Aws, KN, out);
}